// SAModule_37812892074554
// MI455X (gfx1250) — compile-verified
//
#include <hip/hip_runtime.h>
#include <hip/hip_bf16.h>

typedef __attribute__((ext_vector_type(2))) float v2f;
typedef __attribute__((ext_vector_type(8))) float v8f;

#define BATCH 16
#define NPTS  4096
#define NPOINTS 512
#define NSAMPLE 64
#define BN_EPS 1e-5f
#define STAT_BLOCKS 32

// ---------------------------------------------------------------------------
// 1) Farthest-point sampling: one 1024-thread block per batch, coords in LDS.
//    Writes sampled centroid coordinates (B, 512, 3) to `samples`.
// ---------------------------------------------------------------------------
__global__ __launch_bounds__(1024) void fps_kernel(const float* __restrict__ coord,
                                                   float* __restrict__ samples)
{
    __shared__ float sx[NPTS], sy[NPTS], sz[NPTS];
    __shared__ float s_val[32];
    __shared__ int   s_idx[32];
    __shared__ int   s_far;

    const int b   = blockIdx.x;
    const int tid = threadIdx.x;
    const float* cb = coord + (size_t)b * NPTS * 3;

    for (int i = tid; i < NPTS; i += 1024) {
        sx[i] = cb[i * 3 + 0];
        sy[i] = cb[i * 3 + 1];
        sz[i] = cb[i * 3 + 2];
    }
    if (tid == 0) s_far = 0;
    __syncthreads();

    float dist[4];
#pragma unroll
    for (int i = 0; i < 4; i++) dist[i] = 1e10f;

    const int lane = tid & 31;
    const int wave = tid >> 5;

    for (int s = 0; s < NPOINTS; s++) {
        const int far = s_far;
        const float cx = sx[far], cy = sy[far], cz = sz[far];
        if (tid == 0) {
            float* sp = samples + ((size_t)b * NPOINTS + s) * 3;
            sp[0] = cx; sp[1] = cy; sp[2] = cz;
        }
        float bestv = -1.0f; int besti = 0;
#pragma unroll
        for (int i = 0; i < 4; i++) {
            const int p = tid + i * 1024;
            const float dx = sx[p] - cx, dy = sy[p] - cy, dz = sz[p] - cz;
            const float d = dx * dx + dy * dy + dz * dz;
            const float nd = fminf(dist[i], d);
            dist[i] = nd;
            if (nd > bestv || (nd == bestv && p < besti)) { bestv = nd; besti = p; }
        }
        // wave32 argmax (prefer lowest index on ties, matching jnp.argmax)
#pragma unroll
        for (int o = 16; o > 0; o >>= 1) {
            const float ov = __shfl_down(bestv, o, 32);
            const int   oi = __shfl_down(besti, o, 32);
            if (ov > bestv || (ov == bestv && oi < besti)) { bestv = ov; besti = oi; }
        }
        if (lane == 0) { s_val[wave] = bestv; s_idx[wave] = besti; }
        __syncthreads();
        if (tid == 0) {
            float bv = s_val[0]; int bi = s_idx[0];
            for (int w = 1; w < 32; w++) {
                if (s_val[w] > bv || (s_val[w] == bv && s_idx[w] < bi)) {
                    bv = s_val[w]; bi = s_idx[w];
                }
            }
            s_far = bi;
        }
        __syncthreads();
    }
}

// ---------------------------------------------------------------------------
// 2) Ball query: one wave32 per (batch, sample). Ballot-prefix scan keeps the
//    ascending-index order of the reference sort. Pads with first index,
//    or with N when nothing is in range.
// ---------------------------------------------------------------------------
__global__ __launch_bounds__(256) void ballq_kernel(const float* __restrict__ coord,
                                                    const float* __restrict__ samples,
                                                    int* __restrict__ gidx)
{
    const int gw   = (int)((blockIdx.x * blockDim.x + threadIdx.x) >> 5);
    const int lane = threadIdx.x & 31;
    const int b = gw / NPOINTS;
    const int s = gw % NPOINTS;
    if (b >= BATCH) return;

    const float* sp = samples + ((size_t)b * NPOINTS + s) * 3;
    const float cx = sp[0], cy = sp[1], cz = sp[2];
    const float* cb = coord + (size_t)b * NPTS * 3;
    int* out = gidx + ((size_t)b * NPOINTS + s) * NSAMPLE;
    const float r2 = 0.4f * 0.4f;

    int cnt = 0;
    int first = NPTS;
    for (int base = 0; base < NPTS && cnt < NSAMPLE; base += 32) {
        const int j = base + lane;
        const float dx = cb[j * 3 + 0] - cx;
        const float dy = cb[j * 3 + 1] - cy;
        const float dz = cb[j * 3 + 2] - cz;
        const float d = dx * dx + dy * dy + dz * dz;
        const bool in = (d <= r2);
        const unsigned int mask = (unsigned int)__ballot(in);
        if (mask) {
            if (first == NPTS) first = base + __ffs((int)mask) - 1;
            const int rank = __popc(mask & ((1u << lane) - 1u));
            const int pos = cnt + rank;
            if (in && pos < NSAMPLE) out[pos] = j;
            cnt += __popc(mask);
        }
    }
    if (cnt > NSAMPLE) cnt = NSAMPLE;
    for (int p = cnt + lane; p < NSAMPLE; p += 32) out[p] = first;
}

// ---------------------------------------------------------------------------
// 3a) Layer 1 (I=3): scalar, coalesced over n. Output layout (B, C, N).
// ---------------------------------------------------------------------------
__global__ void layer1_kernel(const float* __restrict__ coord,
                              const float* __restrict__ W,
                              const float* __restrict__ bias,
                              float* __restrict__ out)
{
    const int idx = (int)(blockIdx.x * blockDim.x + threadIdx.x);
    const int total = BATCH * 64 * NPTS;
    if (idx >= total) return;
    const int n = idx % NPTS;
    const int c = (idx / NPTS) % 64;
    const int b = idx / (NPTS * 64);
    const float* p = coord + ((size_t)b * NPTS + n) * 3;
    out[idx] = bias[c] + W[c * 3 + 0] * p[0] + W[c * 3 + 1] * p[1] + W[c * 3 + 2] * p[2];
}

// ---------------------------------------------------------------------------
// 3b) WMMA f32 GEMM: Y[b, o, n] = sum_k W[o,k] * H[b,k,n] + bias[o]
//     One 16x16 output tile per wave32, K-loop of V_WMMA_F32_16X16X4_F32.
// ---------------------------------------------------------------------------
__global__ __launch_bounds__(256) void gemm_wmma_kernel(const float* __restrict__ H,
                                                        const float* __restrict__ W,
                                                        const float* __restrict__ bias,
                                                        float* __restrict__ Y,
                                                        int I, int O)
{
    const int lane  = threadIdx.x & 31;
    const int gwave = (int)(blockIdx.x * (blockDim.x >> 5) + (threadIdx.x >> 5));
    const int tilesN = NPTS / 16;
    const int tilesO = O / 16;
    const int tilesPerBatch = tilesO * tilesN;
    const int b = gwave / tilesPerBatch;
    if (b >= BATCH) return;                 // wave-uniform; EXEC stays all-ones
    const int rem = gwave % tilesPerBatch;
    const int o0 = (rem / tilesN) * 16;
    const int n0 = (rem % tilesN) * 16;

    const int hi = lane >> 4;               // half-wave select (K split / row+8)
    const int lo = lane & 15;

    const float* Hb = H + (size_t)b * I * NPTS;
    float*       Yb = Y + (size_t)b * O * NPTS;

    // C/D layout: VGPR i -> row (o0 + i + 8*hi), col (n0 + lo)
    v8f acc;
#pragma unroll
    for (int i = 0; i < 8; i++) acc[i] = bias[o0 + i + 8 * hi];

    for (int k0 = 0; k0 < I; k0 += 4) {
        v2f a, bm;
        // A 16x4: lanes 0-15 rows 0-15 K={k0,k0+1}; lanes 16-31 K={k0+2,k0+3}
        a[0] = W[(o0 + lo) * I + k0 + 2 * hi + 0];
        a[1] = W[(o0 + lo) * I + k0 + 2 * hi + 1];
        // B 4x16: same half-wave K split, col = lo
        bm[0] = Hb[(size_t)(k0 + 2 * hi + 0) * NPTS + n0 + lo];
        bm[1] = Hb[(size_t)(k0 + 2 * hi + 1) * NPTS + n0 + lo];
        acc = __builtin_amdgcn_wmma_f32_16x16x4_f32(
            false, a, false, bm, (short)0, acc, false, false);
    }

#pragma unroll
    for (int i = 0; i < 8; i++)
        Yb[(size_t)(o0 + i + 8 * hi) * NPTS + n0 + lo] = acc[i];
}

// ---------------------------------------------------------------------------
// 3c) BN stats, deterministic two-stage reduction (no float atomics).
//     Stage 1: partial[c*STAT_BLOCKS + bx] = block sum, likewise sumsq.
//     Stage 2: fixed-order reduce of STAT_BLOCKS partials per channel.
// ---------------------------------------------------------------------------
__global__ void stats_partial_kernel(const float* __restrict__ x,
                                     float* __restrict__ partial, int C)
{
    __shared__ float s1[256], s2[256];
    const int c = blockIdx.y;
    const int M = BATCH * NPTS;
    float a = 0.0f, q = 0.0f;
    for (int i = (int)(blockIdx.x * blockDim.x + threadIdx.x); i < M;
         i += (int)(gridDim.x * blockDim.x)) {
        const int b = i / NPTS, n = i % NPTS;
        const float v = x[((size_t)b * C + c) * NPTS + n];
        a += v; q += v * v;
    }
    s1[threadIdx.x] = a; s2[threadIdx.x] = q;
    __syncthreads();
    for (int o = 128; o > 0; o >>= 1) {
        if ((int)threadIdx.x < o) {
            s1[threadIdx.x] += s1[threadIdx.x + o];
            s2[threadIdx.x] += s2[threadIdx.x + o];
        }
        __syncthreads();
    }
    if (threadIdx.x == 0) {
        partial[c * STAT_BLOCKS + blockIdx.x]                    = s1[0];
        partial[(128 + c) * STAT_BLOCKS + blockIdx.x]            = s2[0];
    }
}

__global__ void stats_final_kernel(const float* __restrict__ partial,
                                   float* __restrict__ sums, int C)
{
    const int c = (int)threadIdx.x;
    if (c >= C) return;
    float a = 0.0f, q = 0.0f;
    for (int i = 0; i < STAT_BLOCKS; i++) {
        a += partial[c * STAT_BLOCKS + i];
        q += partial[(128 + c) * STAT_BLOCKS + i];
    }
    sums[c] = a;
    sums[128 + c] = q;
}

// ---------------------------------------------------------------------------
// 3d) In-place batchnorm (training stats) + affine + ReLU.
// ---------------------------------------------------------------------------
__global__ void bn_relu_kernel(float* __restrict__ x, const float* __restrict__ sums,
                               const float* __restrict__ g, const float* __restrict__ be,
                               int C)
{
    const size_t idx = (size_t)blockIdx.x * blockDim.x + threadIdx.x;
    const size_t total = (size_t)BATCH * C * NPTS;
    if (idx >= total) return;
    const int c = (int)((idx / NPTS) % C);
    const float M = (float)(BATCH * NPTS);
    const float mean = sums[c] / M;
    const float var  = sums[128 + c] / M - mean * mean;
    float v = (x[idx] - mean) * rsqrtf(var + BN_EPS);
    v = v * g[c] + be[c];
    x[idx] = fmaxf(v, 0.0f);
}

// ---------------------------------------------------------------------------
// Host-side launch sequence (all on `stream`, graph-capture safe).
// ---------------------------------------------------------------------------
extern "C" void kernel_launch(void* const* d_in, const int* in_sizes, int n_in,
                              void* d_out, int out_size, void* d_ws, size_t ws_size,
                              hipStream_t stream)
{
    (void)in_sizes; (void)n_in; (void)out_size; (void)ws_size;

    const float* coord = (const float*)d_in[0];
    const float* w0 = (const float*)d_in[1];
    const float* b0 = (const float*)d_in[2];
    const float* g0 = (const float*)d_in[3];
    const float* be0 = (const float*)d_in[4];
    const float* w1 = (const float*)d_in[5];
    const float* b1 = (const float*)d_in[6];
    const float* g1 = (const float*)d_in[7];
    const float* be1 = (const float*)d_in[8];
    const float* w2 = (const float*)d_in[9];
    const float* b2 = (const float*)d_in[10];
    const float* g2 = (const float*)d_in[11];
    const float* be2 = (const float*)d_in[12];

    // Output: group_idx (int32, 16*512*64) then feats (float, 16*128*4096)
    int*   gidx  = (int*)d_out;
    float* feats = (float*)d_out + (size_t)BATCH * NPOINTS * NSAMPLE;

    // Workspace layout (byte offsets, 256B-aligned regions)
    char* ws = (char*)d_ws;
    float* samples = (float*)(ws);                 // 16*512*3 floats   (98304 B)
    float* partial = (float*)(ws + 98304);         // 256*32 floats     (32768 B)
    float* sums    = (float*)(ws + 131072);        // 256 floats        (1024 B)
    float* h1      = (float*)(ws + 132096);        // 16*64*4096 floats (16 MiB)
    float* h2      = (float*)(ws + 132096 + (size_t)BATCH * 64 * NPTS * 4);

    // --- sampling / grouping -------------------------------------------------
    fps_kernel<<<BATCH, 1024, 0, stream>>>(coord, samples);

    {
        const int waves = BATCH * NPOINTS;          // one wave32 per sample
        const int blocks = (waves * 32) / 256;
        ballq_kernel<<<blocks, 256, 0, stream>>>(coord, samples, gidx);
    }

    // --- layer 1: 3 -> 64 ----------------------------------------------------
    {
        const int total = BATCH * 64 * NPTS;
        layer1_kernel<<<(total + 255) / 256, 256, 0, stream>>>(coord, w0, b0, h1);
        stats_partial_kernel<<<dim3(STAT_BLOCKS, 64), 256, 0, stream>>>(h1, partial, 64);
        stats_final_kernel<<<1, 128, 0, stream>>>(partial, sums, 64);
        bn_relu_kernel<<<(total + 255) / 256, 256, 0, stream>>>(h1, sums, g0, be0, 64);
    }

    // --- layer 2: 64 -> 64 (WMMA) -------------------------------------------
    {
        const int waves = BATCH * (64 / 16) * (NPTS / 16);
        gemm_wmma_kernel<<<waves / 8, 256, 0, stream>>>(h1, w1, b1, h2, 64, 64);
        const int total = BATCH * 64 * NPTS;
        stats_partial_kernel<<<dim3(STAT_BLOCKS, 64), 256, 0, stream>>>(h2, partial, 64);
        stats_final_kernel<<<1, 128, 0, stream>>>(partial, sums, 64);
        bn_relu_kernel<<<(total + 255) / 256, 256, 0, stream>>>(h2, sums, g1, be1, 64);
    }

    // --- layer 3: 64 -> 128 (WMMA), straight into d_out ----------------------
    {
        const int waves = BATCH * (128 / 16) * (NPTS / 16);
        gemm_wmma_kernel<<<waves / 8, 256, 0, stream>>>(h2, w2, b2, feats, 64, 128);
        const int total = BATCH * 128 * NPTS;
        stats_partial_kernel<<<dim3(STAT_BLOCKS, 128), 256, 0, stream>>>(feats, partial, 128);
        stats_final_kernel<<<1, 128, 0, stream>>>(partial, sums, 128);
        bn_relu_kernel<<<(total + 255) / 256, 256, 0, stream>>>(feats, sums, g2, be2, 128);
    }
}